// PELinear_36412732735663
// MI455X (gfx1250) — compile-verified
//
#include <hip/hip_runtime.h>
#include <hip/hip_bf16.h>

typedef __attribute__((ext_vector_type(2))) float v2f;
typedef __attribute__((ext_vector_type(4))) float v4f;
typedef __attribute__((ext_vector_type(8))) float v8f;

// ---------------------------------------------------------------------------
// Phase 1: per-row reductions.
//   s0[b]  = sum_i x[b,i]
//   s1r[b] = sum_i x[b,i]*w_real[i]
//   s1i[b] = sum_i x[b,i]*w_imag[i]
// One 256-thread workgroup (8 wave32) per row. x is streamed once ->
// non-temporal b128 loads; w stays temporal (L2-resident, 64 KB).
// ---------------------------------------------------------------------------
__global__ void pelinear_reduce_rows(const float* __restrict__ x,
                                     const float* __restrict__ wr,
                                     const float* __restrict__ wi,
                                     float* __restrict__ s,   // [3*Btot]
                                     int N, int Btot) {
    const int row  = blockIdx.x;
    const int nvec = N >> 2;  // float4 count per row
    const v4f* __restrict__ xr  = (const v4f*)(x + (size_t)row * (size_t)N);
    const v4f* __restrict__ wr4 = (const v4f*)wr;
    const v4f* __restrict__ wi4 = (const v4f*)wi;

    float a0 = 0.f, a1 = 0.f, a2 = 0.f;
    for (int c = threadIdx.x; c < nvec; c += blockDim.x) {
        __builtin_prefetch(xr + c + blockDim.x, 0, 0);   // global_prefetch_b8
        v4f xv  = __builtin_nontemporal_load(xr + c);    // stream, don't cache
        v4f wrv = wr4[c];
        v4f wiv = wi4[c];
        a0 += (xv[0] + xv[1]) + (xv[2] + xv[3]);
        a1 += xv[0] * wrv[0] + xv[1] * wrv[1] + xv[2] * wrv[2] + xv[3] * wrv[3];
        a2 += xv[0] * wiv[0] + xv[1] * wiv[1] + xv[2] * wiv[2] + xv[3] * wiv[3];
    }

    // wave32 tree reduction
    for (int off = 16; off > 0; off >>= 1) {
        a0 += __shfl_down(a0, off, 32);
        a1 += __shfl_down(a1, off, 32);
        a2 += __shfl_down(a2, off, 32);
    }

    __shared__ float red0[8], red1[8], red2[8];
    const int waveId = threadIdx.x >> 5;
    const int lane   = threadIdx.x & 31;
    if (lane == 0) { red0[waveId] = a0; red1[waveId] = a1; red2[waveId] = a2; }
    __syncthreads();
    if (threadIdx.x == 0) {
        float t0 = 0.f, t1 = 0.f, t2 = 0.f;
        const int nwaves = blockDim.x >> 5;
        for (int i = 0; i < nwaves; ++i) { t0 += red0[i]; t1 += red1[i]; t2 += red2[i]; }
        s[row]            = t0;   // s0
        s[Btot + row]     = t1;   // s1r
        s[2 * Btot + row] = t2;   // s1i
    }
}

// ---------------------------------------------------------------------------
// Phase 2: rank-2 outer product via V_WMMA_F32_16X16X4_F32.
//   out[m,j].re = (s1r[m]+br) * 1 + s0[m] *  wr[j]
//   out[m,j].im = (s1i[m]+bi) * 1 + s0[m] * (-wi[j])
// A (16x4, K rows 0..1 used): VGPR0 = {K0 | K2}, VGPR1 = {K1 | K3}
// B (4x16):                   VGPR0 = {K0 | K2}, VGPR1 = {K1 | K3}
// D (16x16 f32): VGPR v holds row M = v + 8*(lane>=16), column = lane&15.
// One workgroup = 8 waves per 16-row slab; waves stride over 16-column tiles.
// Branch-free inner loop (cndmask zeroing instead of EXEC divergence);
// interleaved-complex non-temporal b64 stores (two 128B segments each).
// ---------------------------------------------------------------------------
__global__ void pelinear_outer_wmma(const float* __restrict__ wr,
                                    const float* __restrict__ wi,
                                    const float* __restrict__ s,   // [3*Btot]
                                    const float* __restrict__ br,
                                    const float* __restrict__ bi,
                                    float* __restrict__ out,       // interleaved complex
                                    int N, int Btot) {
    const int wave     = threadIdx.x >> 5;
    const int lane     = threadIdx.x & 31;
    const int l15      = lane & 15;
    const bool lowHalf = lane < 16;
    const int rowBase  = blockIdx.x * 16;
    const int colTiles = N >> 4;
    const int nwaves   = blockDim.x >> 5;

    const float breal = br[0];
    const float bimag = bi[0];

    // Build A matrices once per 16-row slab (lanes 16-31 must hold K=2,3 = 0).
    const int m  = rowBase + l15;
    float s0v  = s[m];              // all lanes load (dup addresses hit L2)
    float s1rv = s[Btot + m];
    float s1iv = s[2 * Btot + m];
    v2f Ar, Ai;
    Ar[0] = lowHalf ? (s1rv + breal) : 0.f;  // A[m,0] | A[m,2]=0
    Ar[1] = lowHalf ? s0v            : 0.f;  // A[m,1] | A[m,3]=0
    Ai[0] = lowHalf ? (s1iv + bimag) : 0.f;
    Ai[1] = lowHalf ? s0v            : 0.f;
    const float oneLow = lowHalf ? 1.0f : 0.f;

    const int hi  = lane >> 4;                       // 0 or 1
    const int m0  = rowBase + (hi << 3);             // row base for this half-wave
    const size_t rowStride2 = (size_t)N * 2;         // floats per output row

    for (int jt = wave; jt < colTiles; jt += nwaves) {
        const int j0 = jt << 4;
        // Unconditional loads (high lanes read same addresses, L2 hits),
        // then select-zero for the high half -> no EXEC branch in the loop.
        float wrv = wr[j0 + l15];
        float wiv = wi[j0 + l15];
        wrv = lowHalf ?  wrv : 0.f;   // B[1,n] | B[3,n]=0
        wiv = lowHalf ? -wiv : 0.f;   // conj(w)

        v2f Br, Bi;
        Br[0] = oneLow;               // B[0,n]=1 | B[2,n]=0
        Br[1] = wrv;
        Bi[0] = oneLow;
        Bi[1] = wiv;

        v8f zero = {};
        v8f Dr = __builtin_amdgcn_wmma_f32_16x16x4_f32(
            false, Ar, false, Br, (short)0, zero, false, false);
        v8f Di = __builtin_amdgcn_wmma_f32_16x16x4_f32(
            false, Ai, false, Bi, (short)0, zero, false, false);

        float* op = out + ((size_t)m0 * (size_t)N + (size_t)(j0 + l15)) * 2;
        #pragma unroll
        for (int v = 0; v < 8; ++v) {
            v2f val;
            val[0] = Dr[v];
            val[1] = Di[v];
            // write-once stream: non-temporal global_store_b64
            __builtin_nontemporal_store(val, (v2f*)(op + (size_t)v * rowStride2));
        }
    }
}

extern "C" void kernel_launch(void* const* d_in, const int* in_sizes, int n_in,
                              void* d_out, int out_size, void* d_ws, size_t ws_size,
                              hipStream_t stream) {
    const float* x  = (const float*)d_in[0];
    const float* wr = (const float*)d_in[1];
    const float* wi = (const float*)d_in[2];
    const float* br = (const float*)d_in[3];
    const float* bi = (const float*)d_in[4];

    const int N = in_sizes[1];            // 8192
    const int B = in_sizes[0] / N;        // 8192

    float* out = (float*)d_out;           // interleaved complex64 -> float pairs
    float* s   = (float*)d_ws;            // [3*B] floats: s0 | s1r | s1i

    pelinear_reduce_rows<<<B, 256, 0, stream>>>(x, wr, wi, s, N, B);
    pelinear_outer_wmma<<<B / 16, 256, 0, stream>>>(wr, wi, s, br, bi, out, N, B);
}